// CnnFdmNet_56401510531309
// MI455X (gfx1250) — compile-verified
//
#include <hip/hip_runtime.h>
#include <stdint.h>

// FDTD wave step: u2 = 2*u1 - u0 + r*Lap(u1) - s*(j2 - j0)
//   r = DT^2*C^2/DX^2 = 0.25,  s = DT/(2*EPS) = 0.0025
// B=16, H=W=1024, fp32. Memory-bound: 320 MB -> ~13.7 us floor @ 23.3 TB/s.
//
// Data movement (the only thing that matters at 0.7 FLOP/byte):
//  - u1 tile staged into LDS once per block:
//      * interior blocks: ONE Tensor-Data-Mover descriptor (tensor_load_to_lds,
//        TENSORcnt) DMAs the whole 66x136 fp32 tile, freeing the VMEM pipe.
//      * edge blocks: LDS pre-zeroed (emulates conv zero padding), then
//        bounds-checked per-lane global_load_async_to_lds_b128 (ASYNCcnt).
//  - u0/j2/j0/out are touched exactly once -> non-temporal b128 streams so they
//    don't evict u1 halo lines (reused by neighbor blocks) from L2.

typedef float v4f  __attribute__((ext_vector_type(4)));
typedef unsigned int u32x4 __attribute__((ext_vector_type(4)));
typedef int i32x4 __attribute__((ext_vector_type(4)));
typedef int i32x8 __attribute__((ext_vector_type(8)));

#define TX 128
#define TY 64
#define HALO_X 4                         // 4-float halo so every chunk is b128-aligned
#define LDS_W (TX + 2 * HALO_X)          // 136 floats per LDS row (544 B, 16B-aligned)
#define LDS_H (TY + 2)                   // 66 rows
#define CHUNKS_PER_ROW (LDS_W / 4)       // 34
#define NCHUNK (CHUNKS_PER_ROW * LDS_H)  // 2244 16-byte chunks (~35.9 KB)
#define NTHREADS 256
#define YITER (TY / 8)                   // 8 float4 rows per thread

#if __has_builtin(__builtin_amdgcn_tensor_load_to_lds)
#define HAVE_TDM 1
#else
#define HAVE_TDM 0
#endif

__device__ __forceinline__ void async_chunk_load(const float* gptr, uint32_t lds_off)
{
    asm volatile("global_load_async_to_lds_b128 %0, %1, off"
                 :: "v"(lds_off), "v"(gptr)
                 : "memory");
}

__device__ __forceinline__ void wait_asynccnt0()
{
#if __has_builtin(__builtin_amdgcn_s_wait_asynccnt)
    __builtin_amdgcn_s_wait_asynccnt(0);
#else
    asm volatile("s_wait_asynccnt 0" ::: "memory");
#endif
}

__global__ __launch_bounds__(NTHREADS)
void fdtd_step_kernel(const float* __restrict__ u1, const float* __restrict__ u0,
                      const float* __restrict__ j2, const float* __restrict__ j0,
                      float* __restrict__ out, int H, int W)
{
    __shared__ __align__(16) float tile[LDS_H * LDS_W];

    const int tid = threadIdx.x;
    const int x0  = blockIdx.x * TX;
    const int y0  = blockIdx.y * TY;
    const long long plane = (long long)H * (long long)W;
    const long long ibase = (long long)blockIdx.z * plane;

    const bool edge = (blockIdx.x == 0) | (blockIdx.x == gridDim.x - 1) |
                      (blockIdx.y == 0) | (blockIdx.y == gridDim.y - 1);
    if (edge) {
        // ---- pre-zero LDS so OOB halo == the reference's zero padding ----
        const v4f z = {0.f, 0.f, 0.f, 0.f};
        for (int i = tid; i < NCHUNK; i += NTHREADS) {
            *reinterpret_cast<v4f*>(&tile[i * 4]) = z;
        }
        __syncthreads();  // ds stores complete before async writes land in LDS

        // ---- bounds-checked per-lane async global->LDS b128 staging ----
        for (int i = tid; i < NCHUNK; i += NTHREADS) {
            const int row = i / CHUNKS_PER_ROW;        // gy = y0-1+row
            const int cc  = i - row * CHUNKS_PER_ROW;  // gx = x0-4+4*cc
            const int gy  = y0 - 1 + row;
            const int gx  = x0 - HALO_X + cc * 4;
            if (gy >= 0 && gy < H && gx >= 0 && gx < W) {
                const uint32_t lds_off =
                    (uint32_t)(uintptr_t)(&tile[row * LDS_W + cc * 4]);
                async_chunk_load(u1 + (ibase + (long long)gy * W + gx), lds_off);
            }
        }
        wait_asynccnt0();
    } else {
#if HAVE_TDM
        // ---- interior: single TDM descriptor DMAs the whole tile to LDS ----
        if (tid < 32) {  // one wave issues the DMA (EXEC-independent wave op)
            const uint64_t gaddr =
                (uint64_t)(uintptr_t)(u1 + (ibase + (long long)(y0 - 1) * W + (x0 - HALO_X)));
            const uint32_t lds0 = (uint32_t)(uintptr_t)(&tile[0]);
            const uint32_t tdim0 = (uint32_t)(W - (x0 - HALO_X)); // remaining extent, > tile_dim0
            const uint32_t tdim1 = (uint32_t)(H - (y0 - 1));

            u32x4 g0;
            g0.x = 1u;                                   // count=1, user mode, no gather
            g0.y = lds0;                                 // lds_addr[31:0]
            g0.z = (uint32_t)gaddr;                      // global_addr[31:0]
            g0.w = (uint32_t)((gaddr >> 32) & 0x01FFFFFFu) | 0x80000000u; // addr[56:32] | type=2

            i32x8 g1;
            g1[0] = 0x00020000;                          // wg_mask=0, data_size=2 (4B)
            g1[1] = (int)((tdim0 & 0xFFFFu) << 16);      // tensor_dim0[15:0] (barrier addr=0)
            g1[2] = (int)((tdim0 >> 16) | ((tdim1 & 0xFFFFu) << 16)); // dim0 hi | dim1 lo
            g1[3] = (int)((tdim1 >> 16) | ((uint32_t)LDS_W << 16));   // dim1 hi | tile_dim0=136
            g1[4] = LDS_H;                               // tile_dim1=66, tile_dim2=0
            g1[5] = W;                                   // tensor_dim0_stride[31:0]
            g1[6] = 0;                                   // stride0 hi | stride1 lo
            g1[7] = 0;                                   // stride1 hi

            const i32x4 gz4 = {0, 0, 0, 0};              // 2D tensor: groups 2/3 unused
            const i32x8 gz8 = {0, 0, 0, 0, 0, 0, 0, 0};  // unused trailing group (VADDR4/NULL)
            __builtin_amdgcn_tensor_load_to_lds(g0, g1, gz4, gz4, gz8, 0);
            __builtin_amdgcn_s_wait_tensorcnt(0);
        }
#else
        // ---- interior fallback: per-lane async, no bounds checks needed ----
        for (int i = tid; i < NCHUNK; i += NTHREADS) {
            const int row = i / CHUNKS_PER_ROW;
            const int cc  = i - row * CHUNKS_PER_ROW;
            const uint32_t lds_off =
                (uint32_t)(uintptr_t)(&tile[row * LDS_W + cc * 4]);
            const long long gy = y0 - 1 + row;
            const long long gx = x0 - HALO_X + cc * 4;
            async_chunk_load(u1 + (ibase + gy * W + gx), lds_off);
        }
        wait_asynccnt0();
#endif
    }
    __syncthreads();

    // ---- compute: each thread emits YITER float4 results (rows ry + 8k) ----
    const int xi = (tid & 31) * 4;   // x offset within tile: 0..124
    const int ry = tid >> 5;         // 0..7
    const float r = 0.25f;           // DT^2*C^2/DX^2
    const float s = 0.0025f;         // DT/(2*EPS)

    #pragma unroll
    for (int k = 0; k < YITER; ++k) {
        const int y = ry + k * 8;    // 0..TY-1
        const float* lrow = &tile[(y + 1) * LDS_W + HALO_X + xi];

        const v4f   c  = *reinterpret_cast<const v4f*>(lrow);
        const float lf = lrow[-1];
        const float rt = lrow[4];
        const v4f   up = *reinterpret_cast<const v4f*>(lrow - LDS_W);
        const v4f   dn = *reinterpret_cast<const v4f*>(lrow + LDS_W);

        const long long g = ibase + (long long)(y0 + y) * W + (x0 + xi);
        // streaming operands: non-temporal, keep L2 for u1 halo reuse
        const v4f a0 = __builtin_nontemporal_load(reinterpret_cast<const v4f*>(u0 + g));
        const v4f b2 = __builtin_nontemporal_load(reinterpret_cast<const v4f*>(j2 + g));
        const v4f b0 = __builtin_nontemporal_load(reinterpret_cast<const v4f*>(j0 + g));

        v4f lap;
        lap.x = up.x + dn.x + lf  + c.y - 4.0f * c.x;
        lap.y = up.y + dn.y + c.x + c.z - 4.0f * c.y;
        lap.z = up.z + dn.z + c.y + c.w - 4.0f * c.z;
        lap.w = up.w + dn.w + c.z + rt  - 4.0f * c.w;

        v4f o;
        o.x = 2.0f * c.x - a0.x + r * lap.x - s * (b2.x - b0.x);
        o.y = 2.0f * c.y - a0.y + r * lap.y - s * (b2.y - b0.y);
        o.z = 2.0f * c.z - a0.z + r * lap.z - s * (b2.z - b0.z);
        o.w = 2.0f * c.w - a0.w + r * lap.w - s * (b2.w - b0.w);

        __builtin_nontemporal_store(o, reinterpret_cast<v4f*>(out + g));
    }
}

extern "C" void kernel_launch(void* const* d_in, const int* in_sizes, int n_in,
                              void* d_out, int out_size, void* d_ws, size_t ws_size,
                              hipStream_t stream)
{
    (void)n_in; (void)out_size; (void)d_ws; (void)ws_size;
    const float* u1 = (const float*)d_in[0];
    const float* u0 = (const float*)d_in[1];
    const float* j2 = (const float*)d_in[2];
    const float* j0 = (const float*)d_in[3];
    float* out = (float*)d_out;

    const int H = 1024, W = 1024;
    const int B = in_sizes[0] / (H * W);   // 16

    dim3 grid(W / TX, H / TY, B);          // (8, 16, 16)
    dim3 block(NTHREADS);
    fdtd_step_kernel<<<grid, block, 0, stream>>>(u1, u0, j2, j0, out, H, W);
}